// YoloV1Loss_24257975288348
// MI455X (gfx1250) — compile-verified
//
#include <hip/hip_runtime.h>
#include <hip/hip_bf16.h>
#include <stdint.h>

#define LAMBDA_COORD 5.0f
#define LAMBDA_NOOBJ 0.5f
#define CELL (1.0f / 7.0f)

typedef __attribute__((ext_vector_type(2))) float v2f;
typedef __attribute__((ext_vector_type(8))) float v8f;

// ---------------------------------------------------------------------------
// Async global -> LDS copy of 16 bytes (CDNA5 GLOBAL_LOAD_ASYNC_TO_LDS_B128).
// Inline asm so it works on both ROCm 7.2 (clang-22) and amdgpu-toolchain
// (clang-23) regardless of builtin arity differences.
// ---------------------------------------------------------------------------
__device__ __forceinline__ void async_cp_b128(const float* gsrc, float* ldst) {
    unsigned lds_off = (unsigned)(uintptr_t)ldst;          // low 32 bits = LDS offset
    unsigned long long gaddr = (unsigned long long)(uintptr_t)gsrc;
    asm volatile("global_load_async_to_lds_b128 %0, %1, off"
                 :: "v"(lds_off), "v"(gaddr) : "memory");
}

__device__ __forceinline__ void wait_async_zero() {
    asm volatile("s_wait_asynccnt 0x0" ::: "memory");
}

// ---------------------------------------------------------------------------
// Kernel 1: per-block object count and noobj-loss partial.
// ---------------------------------------------------------------------------
__global__ void k_count(const float* __restrict__ P, const float* __restrict__ T,
                        unsigned* __restrict__ blockCount,
                        float* __restrict__ blockNoobj, long long N) {
    __shared__ unsigned sc[256];
    __shared__ float    sn[256];
    int tid = threadIdx.x;
    long long cell = (long long)blockIdx.x * 256 + tid;
    unsigned m = 0u;
    float nb = 0.0f;
    if (cell < N) {
        const float* t = T + cell * 30;
        const float* p = P + cell * 30;
        float conf = t[4];
        m = (conf == 1.0f) ? 1u : 0u;
        if (conf == 0.0f) {
            float d0 = p[4] - t[4];
            float d1 = p[9] - t[9];
            nb = d0 * d0 + d1 * d1;
        }
    }
    sc[tid] = m; sn[tid] = nb;
    __syncthreads();
    for (int off = 128; off > 0; off >>= 1) {
        if (tid < off) { sc[tid] += sc[tid + off]; sn[tid] += sn[tid + off]; }
        __syncthreads();
    }
    if (tid == 0) { blockCount[blockIdx.x] = sc[0]; blockNoobj[blockIdx.x] = sn[0]; }
}

// ---------------------------------------------------------------------------
// Kernel 2: single-block exclusive scan of block counts -> block rank offsets,
// n_obj/2, and deterministic noobj-loss total.
// ---------------------------------------------------------------------------
__global__ void k_scan(const unsigned* __restrict__ cnt,
                       const float* __restrict__ nb,
                       unsigned* __restrict__ offs,
                       unsigned* __restrict__ halfPtr,
                       float* __restrict__ noobjTotal, int nblocks) {
    __shared__ unsigned su[256];
    __shared__ float    sf[256];
    int t = threadIdx.x;
    int K = (nblocks + 255) / 256;
    int s0 = t * K;
    int s1 = s0 + K; if (s1 > nblocks) s1 = nblocks;
    unsigned lu = 0u; float lf = 0.0f;
    for (int i = s0; i < s1; ++i) { lu += cnt[i]; lf += nb[i]; }
    su[t] = lu; sf[t] = lf;
    __syncthreads();
    for (int off = 1; off < 256; off <<= 1) {
        unsigned u = (t >= off) ? su[t - off] : 0u;
        float    f = (t >= off) ? sf[t - off] : 0.0f;
        __syncthreads();
        su[t] += u; sf[t] += f;
        __syncthreads();
    }
    unsigned run = (t == 0) ? 0u : su[t - 1];   // exclusive prefix of this chunk
    for (int i = s0; i < s1; ++i) { offs[i] = run; run += cnt[i]; }
    if (t == 255) {
        *halfPtr    = su[255] / 2u;   // n_obj // 2
        *noobjTotal = sf[255];
    }
}

// ---------------------------------------------------------------------------
// Kernel 3: main loss pass. Async-stage block tiles of P and T into LDS,
// per-cell IoU/argmax/loss, intra-block scan for rank, block reduction.
// 256 cells/block; tile = 256*30 floats = 30720 B per array.
// ---------------------------------------------------------------------------
__global__ void k_loss(const float* __restrict__ P, const float* __restrict__ T,
                       const unsigned* __restrict__ blockOffset,
                       const unsigned* __restrict__ halfPtr,
                       float* __restrict__ blockLoss, long long N) {
    __shared__ float sP[7680];
    __shared__ float sT[7680];
    __shared__ int   sScan[256];
    __shared__ float sRed[256];

    int tid = threadIdx.x;
    int b   = blockIdx.x;
    long long base = (long long)b * 256;
    const float* gP = P + base * 30;
    const float* gT = T + base * 30;

    long long remFloats = (N - base) * 30;
    int nv = (int)(remFloats / 4); if (nv > 1920) nv = 1920;

    for (int i = tid; i < nv; i += 256) async_cp_b128(gP + i * 4, &sP[i * 4]);
    for (int i = tid; i < nv; i += 256) async_cp_b128(gT + i * 4, &sT[i * 4]);
    wait_async_zero();
    __syncthreads();

    long long cell = base + tid;
    int   m = 0;
    float cterm = 0.0f;
    if (cell < N) {
        const float* p = &sP[tid * 30];
        const float* t = &sT[tid * 30];
        float conf = t[4];
        m = (conf == 1.0f) ? 1 : 0;

        // target box
        float tb0 = t[0] * t[0], tb1 = t[1] * t[1];
        float tb2 = t[2] * t[2], tb3 = t[3] * t[3];
        float tax = tb0 * CELL - tb2, tay = tb1 * CELL - tb3;
        float tbx = tax * CELL + tb2, tby = tay * CELL + tb3;
        float areaT = (tbx - tax) * (tby - tay);

        float iou0 = 0.0f, iou1 = 0.0f;
        #pragma unroll
        for (int j = 0; j < 2; ++j) {
            float px = p[5 * j + 0], py = p[5 * j + 1];
            float pw = p[5 * j + 2], ph = p[5 * j + 3];
            float pax = px * CELL - pw, pay = py * CELL - ph;
            float pbx = pax * CELL + pw, pby = pay * CELL + ph;
            float ltx = fmaxf(pax, tax), lty = fmaxf(pay, tay);
            float rbx = fminf(pbx, tbx), rby = fminf(pby, tby);
            float wx = fmaxf(rbx - ltx, 0.0f), wy = fmaxf(rby - lty, 0.0f);
            float inter = wx * wy;
            float areaP = (pbx - pax) * (pby - pay);
            float iou = inter / (areaP + areaT - inter);
            if (j == 0) iou0 = iou; else iou1 = iou;
        }
        int idx = (iou1 > iou0) ? 1 : 0;   // argmax, first on tie
        float s0 = p[5 * idx + 0], s1 = p[5 * idx + 1];
        float s2 = p[5 * idx + 2], s3 = p[5 * idx + 3];

        // class argmax over t[10:30], first occurrence of max
        float best = t[10]; int bi = 0;
        #pragma unroll
        for (int j = 1; j < 20; ++j) {
            float v = t[10 + j];
            if (v > best) { best = v; bi = j; }
        }
        float selc = p[10 + bi];

        float d0 = s0 - t[0], d1 = s1 - t[1];
        float d2 = s2 - t[2], d3 = s3 - t[3];
        float dc = selc - 1.0f;
        cterm = d0 * d0 + d1 * d1 + d2 * d2 + d3 * d3 + 2.0f * dc * dc;
    }

    // intra-block inclusive scan of mask -> rank
    sScan[tid] = m;
    __syncthreads();
    for (int off = 1; off < 256; off <<= 1) {
        int v = (tid >= off) ? sScan[tid - off] : 0;
        __syncthreads();
        sScan[tid] += v;
        __syncthreads();
    }
    unsigned rank = blockOffset[b] + (unsigned)sScan[tid]; // inclusive global rank
    unsigned half = *halfPtr;
    float w = (m && rank <= half) ? 1.0f : 0.0f;

    sRed[tid] = w * cterm;
    __syncthreads();
    for (int off = 128; off > 0; off >>= 1) {
        if (tid < off) sRed[tid] += sRed[tid + off];
        __syncthreads();
    }
    if (tid == 0) blockLoss[b] = sRed[0];
}

// ---------------------------------------------------------------------------
// Kernel 4: single wave32. Sum 3136 block partials (fixed order), collapse the
// 32 lane partials with V_WMMA_F32_16X16X4_F32 (B = ones => row sums), write
// the final scalar loss.
// ---------------------------------------------------------------------------
__global__ void k_final(const float* __restrict__ blockLoss,
                        const float* __restrict__ noobjTotal,
                        float* __restrict__ out, int nblocks) {
    __shared__ float sTmp[32];
    int lane = threadIdx.x;     // 32 threads exactly; EXEC all ones for WMMA
    float acc = 0.0f;
    for (int i = lane; i < nblocks; i += 32) acc += blockLoss[i];

    // A (16x4 f32): lanes 0-15 VGPR0=K0, VGPR1=K1; lanes 16-31 VGPR0=K2, VGPR1=K3.
    // With B = all ones, D[m][n] = A[m][0]+A[m][2] = acc(lane m) + acc(lane m+16).
    v2f a; a[0] = acc;  a[1] = 0.0f;
    v2f bm; bm[0] = 1.0f; bm[1] = 1.0f;
    v8f c = {0.0f, 0.0f, 0.0f, 0.0f, 0.0f, 0.0f, 0.0f, 0.0f};
    v8f d = __builtin_amdgcn_wmma_f32_16x16x4_f32(false, a, false, bm,
                                                  (short)0, c, false, false);
    // lanes 0-15 hold rows 0-7 in their 8 D VGPRs, lanes 16-31 hold rows 8-15.
    float s = d[0] + d[1] + d[2] + d[3] + d[4] + d[5] + d[6] + d[7];
    sTmp[lane] = s;
    __syncthreads();
    if (lane == 0)
        out[0] = LAMBDA_COORD * (sTmp[0] + sTmp[16]) + LAMBDA_NOOBJ * noobjTotal[0];
}

// ---------------------------------------------------------------------------
extern "C" void kernel_launch(void* const* d_in, const int* in_sizes, int n_in,
                              void* d_out, int out_size, void* d_ws, size_t ws_size,
                              hipStream_t stream) {
    const float* P = (const float*)d_in[0];   // predictions (B,S,30) f32
    const float* T = (const float*)d_in[1];   // targets     (B,S,30) f32
    float* out = (float*)d_out;

    long long N = (long long)in_sizes[0] / 30;          // number of cells (802816)
    int nblocks = (int)((N + 255) / 256);               // 3136

    char* ws = (char*)d_ws;
    size_t o = 0;
    unsigned* blockCount  = (unsigned*)(ws + o); o += (size_t)nblocks * 4;
    unsigned* blockOffset = (unsigned*)(ws + o); o += (size_t)nblocks * 4;
    float*    blockNoobj  = (float*)(ws + o);    o += (size_t)nblocks * 4;
    float*    blockLoss   = (float*)(ws + o);    o += (size_t)nblocks * 4;
    unsigned* halfPtr     = (unsigned*)(ws + o); o += 4;
    float*    noobjTotal  = (float*)(ws + o);    o += 4;

    k_count<<<nblocks, 256, 0, stream>>>(P, T, blockCount, blockNoobj, N);
    k_scan <<<1, 256, 0, stream>>>(blockCount, blockNoobj, blockOffset,
                                   halfPtr, noobjTotal, nblocks);
    k_loss <<<nblocks, 256, 0, stream>>>(P, T, blockOffset, halfPtr, blockLoss, N);
    k_final<<<1, 32, 0, stream>>>(blockLoss, noobjTotal, out, nblocks);
}